// MCFNN_82523501625649
// MI455X (gfx1250) — compile-verified
//
#include <hip/hip_runtime.h>
#include <hip/hip_bf16.h>
#include <math.h>

typedef float v2f __attribute__((ext_vector_type(2)));
typedef float v8f __attribute__((ext_vector_type(8)));

#define DIMX 64      // feature dim of x and of sim_emb
#define DIME 384     // feature dim of emb
#define THREADS 256
#define JTILE 32     // j rows staged per LDS tile
#define JSPLIT 8     // split of the j-range across blocks (occupancy)

// ---------------------------------------------------------------------------
// Kernel 1: sim = emb @ W + b using V_WMMA_F32_16X16X4_F32 (exact f32 GEMM).
// One wave per 16x16 output tile. K=384 -> 96 WMMA issues.
// A frag (16x4 f32, 2 VGPR): lane L holds M=L&15, K = 2*(L>>4) + {0,1}.
// B frag (4x16 f32, 2 VGPR): lane L holds N=L&15, K = 2*(L>>4) + {0,1}.
// C/D (16x16 f32, 8 VGPR): VGPR r = row M = r + 8*(L>>4), col N = L&15.
// ---------------------------------------------------------------------------
__global__ __launch_bounds__(32) void mi_gemm_wmma(
    const float* __restrict__ emb, const float* __restrict__ W,
    const float* __restrict__ bias, float* __restrict__ sim)
{
    const int tm   = blockIdx.x * 16;
    const int tn   = blockIdx.y * 16;
    const int lane = threadIdx.x;
    const int hi   = lane >> 4;    // 0 or 1
    const int l15  = lane & 15;

    const float* arow = emb + (size_t)(tm + l15) * DIME + 2 * hi; // A: row M, K offset
    const float* bcol = W + (size_t)(2 * hi) * DIMX + (tn + l15); // B: col N, K offset

    v8f acc = {};
    for (int k = 0; k < DIME; k += 4) {
        v2f a; a.x = arow[k];          a.y = arow[k + 1];
        v2f b; b.x = bcol[k * DIMX];   b.y = bcol[k * DIMX + DIMX];
        acc = __builtin_amdgcn_wmma_f32_16x16x4_f32(
            false, a, false, b, (short)0, acc, false, false);
    }

    const float bv = bias[tn + l15];
    float* dcol = sim + tn + l15;
#pragma unroll
    for (int r = 0; r < 8; ++r)
        dcol[(size_t)(tm + r + 8 * hi) * DIMX] = acc[r] + bv;
}

// Branchless insert of v into descending top-4 (7 min/max ops).
__device__ inline void ins4(float (&t)[4], float v) {
    float m0 = fmaxf(t[0], v); float r = fminf(t[0], v);
    float m1 = fmaxf(t[1], r); r = fminf(t[1], r);
    float m2 = fmaxf(t[2], r); r = fminf(t[2], r);
    float m3 = fmaxf(t[3], r);
    t[0] = m0; t[1] = m1; t[2] = m2; t[3] = m3;
}

// ---------------------------------------------------------------------------
// Kernel 2: streaming pairwise Chebyshev. Thread owns row i (128 f32 in VGPRs),
// j-rows staged through LDS in 16KB tiles (broadcast ds_load_b128).
// Produces per (j-slice, i): rowmax_xy, top4(D_x), top4(D_y).
// ---------------------------------------------------------------------------
__global__ __launch_bounds__(THREADS) void mi_cheb_partial(
    const float* __restrict__ x, const float* __restrict__ sim,
    float* __restrict__ part, int Nn)
{
    __shared__ float4 tile[JTILE * 32];   // 32 float4 = 128 floats per row

    const int i  = blockIdx.x * THREADS + threadIdx.x;
    const int jq = blockIdx.y;
    const int jchunk = Nn / JSPLIT;

    // Own row: dims 0..63 from x, 64..127 from sim (the "concat").
    float4 xi[32];
    const float4* xr = (const float4*)(x   + (size_t)i * DIMX);
    const float4* sr = (const float4*)(sim + (size_t)i * DIMX);
#pragma unroll
    for (int c = 0; c < 16; ++c) xi[c]      = xr[c];
#pragma unroll
    for (int c = 0; c < 16; ++c) xi[16 + c] = sr[c];

    float tx[4] = {-1.f, -1.f, -1.f, -1.f};
    float ty[4] = {-1.f, -1.f, -1.f, -1.f};
    float mxy = -1.f;

    const int j0beg = jq * jchunk;
    for (int j0 = j0beg; j0 < j0beg + jchunk; j0 += JTILE) {
        __syncthreads();
        for (int t = threadIdx.x; t < JTILE * 32; t += THREADS) {
            const int jr = t >> 5, c = t & 31;
            const int j  = j0 + jr;
            tile[t] = (c < 16) ? ((const float4*)(x   + (size_t)j * DIMX))[c]
                               : ((const float4*)(sim + (size_t)j * DIMX))[c - 16];
        }
        __syncthreads();

        for (int jr = 0; jr < JTILE; ++jr) {
            const float4* row = &tile[jr * 32];
            float dx = 0.f, dy = 0.f;
#pragma unroll
            for (int c = 0; c < 16; ++c) {
                float4 v = row[c];
                dx = fmaxf(dx, fabsf(xi[c].x - v.x));
                dx = fmaxf(dx, fabsf(xi[c].y - v.y));
                dx = fmaxf(dx, fabsf(xi[c].z - v.z));
                dx = fmaxf(dx, fabsf(xi[c].w - v.w));
            }
#pragma unroll
            for (int c = 0; c < 16; ++c) {
                float4 v = row[16 + c];
                dy = fmaxf(dy, fabsf(xi[16 + c].x - v.x));
                dy = fmaxf(dy, fabsf(xi[16 + c].y - v.y));
                dy = fmaxf(dy, fabsf(xi[16 + c].z - v.z));
                dy = fmaxf(dy, fabsf(xi[16 + c].w - v.w));
            }
            mxy = fmaxf(mxy, fmaxf(dx, dy));
            ins4(tx, dx);
            ins4(ty, dy);
        }
    }

    float* p = part + ((size_t)jq * Nn + i) * 9;
    p[0] = mxy;
    p[1] = tx[0]; p[2] = tx[1]; p[3] = tx[2]; p[4] = tx[3];
    p[5] = ty[0]; p[6] = ty[1]; p[7] = ty[2]; p[8] = ty[3];
}

// ---------------------------------------------------------------------------
// Kernel 3: merge the JSPLIT partials per row, derive counts, histogram them
// with integer atomics (order-independent => deterministic).
// cnt[0..4] bins count(top4_x <= rowmax_xy) ; cnt[5..9] same for y.
// ---------------------------------------------------------------------------
__global__ __launch_bounds__(256) void mi_merge(
    const float* __restrict__ part, int* __restrict__ cnt, int Nn)
{
    const int i = blockIdx.x * 256 + threadIdx.x;
    float mxy = -1.f;
    float tx[4] = {-1.f, -1.f, -1.f, -1.f};
    float ty[4] = {-1.f, -1.f, -1.f, -1.f};

    for (int q = 0; q < JSPLIT; ++q) {
        const float* p = part + ((size_t)q * Nn + i) * 9;
        mxy = fmaxf(mxy, p[0]);
        ins4(tx, p[1]); ins4(tx, p[2]); ins4(tx, p[3]); ins4(tx, p[4]);
        ins4(ty, p[5]); ins4(ty, p[6]); ins4(ty, p[7]); ins4(ty, p[8]);
    }
    int cx = (tx[0] <= mxy) + (tx[1] <= mxy) + (tx[2] <= mxy) + (tx[3] <= mxy);
    int cy = (ty[0] <= mxy) + (ty[1] <= mxy) + (ty[2] <= mxy) + (ty[3] <= mxy);
    atomicAdd(&cnt[cx], 1);
    atomicAdd(&cnt[5 + cy], 1);
}

// ---------------------------------------------------------------------------
// Kernel 4: epilogue.  mi = log N + log k - mean(log(n_x+1) + log(n_y+1)),
// where log(n_x+1) = log(bin index).
// ---------------------------------------------------------------------------
__global__ void mi_final(const int* __restrict__ cnt, const int* __restrict__ kp,
                         float* __restrict__ out, int Nn)
{
    float s = 0.f;
    for (int v = 1; v <= 4; ++v) {
        const int c = cnt[v] + cnt[5 + v];
        if (c > 0) s += (float)c * logf((float)v);
    }
    if (cnt[0] + cnt[5] > 0) s = -INFINITY;  // log(0) rows
    out[0] = logf((float)Nn) + logf((float)kp[0]) - s / (float)Nn;
}

extern "C" void kernel_launch(void* const* d_in, const int* in_sizes, int n_in,
                              void* d_out, int out_size, void* d_ws, size_t ws_size,
                              hipStream_t stream)
{
    const float* x    = (const float*)d_in[0];
    const float* emb  = (const float*)d_in[1];
    const float* W    = (const float*)d_in[2];
    const float* bias = (const float*)d_in[3];
    const int*   kp   = (const int*)d_in[4];
    const int Nn = in_sizes[0] / DIMX;   // 4096

    float* sim = (float*)d_ws;
    const size_t simBytes  = (size_t)Nn * DIMX * sizeof(float);
    float* part = (float*)((char*)d_ws + simBytes);
    const size_t partBytes = (size_t)JSPLIT * Nn * 9 * sizeof(float);
    int* cnt = (int*)((char*)d_ws + simBytes + partBytes);

    hipMemsetAsync(cnt, 0, 10 * sizeof(int), stream);
    mi_gemm_wmma<<<dim3(Nn / 16, DIMX / 16), 32, 0, stream>>>(emb, W, bias, sim);
    mi_cheb_partial<<<dim3(Nn / THREADS, JSPLIT), THREADS, 0, stream>>>(x, sim, part, Nn);
    mi_merge<<<dim3(Nn / 256), 256, 0, stream>>>(part, cnt, Nn);
    mi_final<<<1, 1, 0, stream>>>(cnt, kp, (float*)d_out, Nn);
}